// CTCLayer_65584150610302
// MI455X (gfx1250) — compile-verified
//
#include <hip/hip_runtime.h>
#include <stdint.h>

// Problem constants (match reference: B=256, T=256, C=512, L=64)
#define BB    256
#define TT    256
#define CC    512
#define LL    64
#define SS    129            // 2*L+1 extended states
#define BLANKC (CC - 1)
#define ROWS  8              // timesteps per TDM tile (8 * 512 * 4B = 16 KiB)
#define NCH   (TT / ROWS)
#define NEGF  (-1e30f)
#define EPSF  (1e-7f)

typedef __attribute__((ext_vector_type(4))) unsigned int u32x4;
typedef __attribute__((ext_vector_type(8))) int          i32x8;
typedef __attribute__((ext_vector_type(4))) int          i32x4;

// Issue one TDM tensor_load_to_lds: 2-D tile (tile_dim0=C floats, tile_dim1=nrows)
// from global row-major [rows, C] at gsrc into LDS at byte offset lds_off.
// D# layout per CDNA5 ISA 8.3/8.4: group0 = {flags, lds_addr, global_addr, type=2},
// group1 = {data_size=4B, tensor_dim0=C, tensor_dim1=rows_left, tile dims, stride=C}.
// This toolchain exposes the 6-arg builtin (extra int32x8 descriptor group);
// our tensor is 2-D so the trailing groups carry no live fields.
__device__ __forceinline__ void tdm_load_rows(const float* gsrc, unsigned lds_off,
                                              int nrows, int rows_left) {
  unsigned long long ga = (unsigned long long)(uintptr_t)gsrc;
  u32x4 g0;
  g0[0] = 1u;                                             // count=1 (valid user D#)
  g0[1] = lds_off;                                        // lds_addr (bytes)
  g0[2] = (unsigned)(ga & 0xFFFFFFFFull);                 // global_addr[31:0]
  g0[3] = (unsigned)((ga >> 32) & 0x01FFFFFFull)          // global_addr[56:32]
        | (2u << 30);                                     // type=2 ("image")
  i32x8 g1;
  g1[0] = (int)(2u << 16);                                // data_size=2 -> 4 bytes
  g1[1] = (int)(((unsigned)CC & 0xFFFFu) << 16);          // tensor_dim0[15:0]=512
  g1[2] = (int)(((unsigned)rows_left & 0xFFFFu) << 16);   // tensor_dim1[15:0]
  g1[3] = (int)(((unsigned)CC & 0xFFFFu) << 16);          // tile_dim0=512
  g1[4] = (int)((unsigned)nrows & 0xFFFFu);               // tile_dim1=nrows, tile_dim2=0
  g1[5] = CC;                                             // tensor_dim0_stride[31:0]=512
  g1[6] = 0;                                              // stride hi / dim1_stride
  g1[7] = 0;
  i32x4 g2 = {0, 0, 0, 0};                                // 2-D tensor: groups 2/3 unused
  i32x4 g3 = {0, 0, 0, 0};
  i32x8 g4 = {0, 0, 0, 0, 0, 0, 0, 0};                    // extra group (clang-23 form)
  __builtin_amdgcn_tensor_load_to_lds(g0, g1, g2, g3, g4, 0);
}

__global__ void __launch_bounds__(160)
ctc_alpha_kernel(const int* __restrict__ y_true,   // [B, L] int (0 = pad)
                 const float* __restrict__ y_pred, // [B, T, C] softmax probs
                 float* __restrict__ out) {        // [B, 1] nll
  __shared__ float buf[2][ROWS * CC];   // double-buffered TDM landing zone (32 KiB)
  __shared__ float alpha[2][SS];        // ping-pong alpha
  __shared__ int   exts[SS];
  __shared__ int   skips[SS];
  __shared__ int   llen;

  const int b   = blockIdx.x;
  const int tid = threadIdx.x;

  // Build extended label sequence: blank at even s, label[(s-1)/2] at odd s.
  if (tid < SS) {
    int e = BLANKC;
    if (tid & 1) e = y_true[b * LL + (tid >> 1)];
    exts[tid] = e;
  }
  if (tid == 0) {
    int c = 0;
    for (int i = 0; i < LL; ++i) c += (y_true[b * LL + i] != 0) ? 1 : 0;
    llen = c;
  }
  __syncthreads();
  if (tid < SS) {
    skips[tid] = (tid >= 2) && (exts[tid] != BLANKC) && (exts[tid] != exts[tid - 2]);
  }

  const float*   base   = y_pred + (size_t)b * TT * CC;
  const unsigned lds_b0 = (unsigned)(uintptr_t)&buf[0][0];
  const unsigned lds_b1 = (unsigned)(uintptr_t)&buf[1][0];
  const bool     issuer = (tid < 32);   // wave 0 owns TDM issue + tensorcnt

  // Preload chunk 0.
  if (issuer) {
    tdm_load_rows(base, lds_b0, ROWS, TT);
    __builtin_amdgcn_s_wait_tensorcnt(0);
  }
  __syncthreads();

  // t = 0 init: alpha0[0] = logp(blank), alpha0[1] = logp(ext[1]), else -inf.
  int cur = 0;
  if (tid < SS) {
    float a = NEGF;
    if (tid < 2) a = __logf(buf[0][exts[tid]] + EPSF);
    alpha[0][tid] = a;
  }
  __syncthreads();

  for (int k = 0; k < NCH; ++k) {
    const int bk = k & 1;
    // Kick off next tile while we compute this one (TENSORcnt in flight).
    if (issuer && (k + 1 < NCH)) {
      tdm_load_rows(base + (size_t)(k + 1) * ROWS * CC,
                    ((k + 1) & 1) ? lds_b1 : lds_b0,
                    ROWS, TT - (k + 1) * ROWS);
    }
    for (int r = (k == 0 ? 1 : 0); r < ROWS; ++r) {
      float nv = NEGF;
      if (tid < SS) {
        float a0 = alpha[cur][tid];
        float a1 = (tid >= 1) ? alpha[cur][tid - 1] : NEGF;
        float a2 = skips[tid] ? alpha[cur][tid - 2] : NEGF;
        float m  = fmaxf(a0, fmaxf(a1, a2));
        float s  = __expf(a0 - m) + __expf(a1 - m) + __expf(a2 - m);
        float lp = __logf(buf[bk][r * CC + exts[tid]] + EPSF);
        nv = m + __logf(s) + lp;
      }
      if (tid < SS) alpha[cur ^ 1][tid] = nv;   // write to the *other* buffer
      cur ^= 1;
      __syncthreads();                          // one barrier per timestep
    }
    // Next tile must have landed before anyone reads it.
    if (issuer) __builtin_amdgcn_s_wait_tensorcnt(0);
    __syncthreads();
  }

  if (tid == 0) {
    int   idx = 2 * llen;                       // final blank state
    float aL  = alpha[cur][idx];
    float aP  = alpha[cur][idx - 1];
    float m   = fmaxf(aL, aP);
    out[b] = -(m + __logf(__expf(aL - m) + __expf(aP - m)));
  }
}

extern "C" void kernel_launch(void* const* d_in, const int* in_sizes, int n_in,
                              void* d_out, int out_size, void* d_ws, size_t ws_size,
                              hipStream_t stream) {
  (void)in_sizes; (void)n_in; (void)out_size; (void)d_ws; (void)ws_size;
  const int*   y_true = (const int*)d_in[0];    // [B, L]
  const float* y_pred = (const float*)d_in[1];  // [B, T, C]
  float*       out    = (float*)d_out;          // [B, 1]
  ctc_alpha_kernel<<<BB, 160, 0, stream>>>(y_true, y_pred, out);
}